// GCNExtractor_4217657885056
// MI455X (gfx1250) — compile-verified
//
#include <hip/hip_runtime.h>
#include <hip/hip_bf16.h>

// Problem constants (fixed by the reference).
#define BB   8
#define NN   20000
#define EE   320000
#define DD   128            // feature dim (= H)
#define BN   (BB*NN)        // 160000 rows total
#define LEAKY 0.01f

typedef __attribute__((ext_vector_type(2))) float v2f;
typedef __attribute__((ext_vector_type(8))) float v8f;

// ---------------------------------------------------------------------------
// K1: deg = 1 (self loop)
__global__ void k_deg_init(float* deg) {
    int i = blockIdx.x * 256 + threadIdx.x;
    if (i < BN) deg[i] = 1.0f;
}

// K2: deg[b, dst] += 1 for every edge
__global__ void k_deg_accum(const long long* __restrict__ ei, float* __restrict__ deg) {
    long long t = (long long)blockIdx.x * 256 + threadIdx.x;
    if (t >= (long long)BB * EE) return;
    int b = (int)(t / EE);
    int e = (int)(t % EE);
    int d = (int)ei[(size_t)b * 2 * EE + EE + e];
    atomicAdd(&deg[b * NN + d], 1.0f);
}

// K3: in-place deg -> dinv = rsqrt(deg)
__global__ void k_dinv(float* deg) {
    int i = blockIdx.x * 256 + threadIdx.x;
    if (i < BN) {
        float d = deg[i];
        deg[i] = (d > 0.0f) ? rsqrtf(d) : 0.0f;
    }
}

// ---------------------------------------------------------------------------
// Tiled WMMA GEMM: Out[Mtot,128] = In[Mtot,128] * W[128,128] (+bias)(+leaky)
// Block = 256 threads = 8 waves. Each block handles TPB=16 consecutive 16-row
// M tiles (256 rows): the 64KB weight is staged into LDS ONCE per block and
// the 8KB A-strip is double-buffered so tile t+1's global loads overlap tile
// t's 32-deep v_wmma_f32_16x16x4_f32 chain. Wave w owns output cols [16w,16w+16).
// f32 WMMA fragment layout (ISA 7.12.2):
//   A (16x4, 2 VGPR): lane<16 -> M=lane, K={k0,k0+1}; lane>=16 -> K={k0+2,k0+3}
//   B (4x16, 2 VGPR): mirrored (row k striped over 16 lanes per half)
//   C (16x16, 8 VGPR): vgpr j: lane<16 -> M=j, N=lane; lane>=16 -> M=j+8
#define LDS_PITCH 132
#define TPB 16   // 16-row M tiles per block
__global__ __launch_bounds__(256) void k_gemm128(
    const float* __restrict__ A0,   // concat: node base | else: input buffer
    const float* __restrict__ A1,   // concat: actions base | else unused
    const float* __restrict__ W,    // [128,128] row-major [in][out]
    const float* __restrict__ bias, // [128] or nullptr
    float* __restrict__ Out,        // [Mtot,128]
    int concat_mode, int act_leaky)
{
    __shared__ float Ws[128 * LDS_PITCH];     // 67584 B
    __shared__ float Xs[2][16 * LDS_PITCH];   // 2 x 8448 B
    const int tid   = threadIdx.x;
    const int tile0 = blockIdx.x * TPB;

    // Stage W once: 4096 float4, 16 per thread.
    for (int i = tid; i < 128 * 32; i += 256) {
        int k  = i >> 5;
        int n4 = i & 31;
        float4 v = ((const float4*)W)[i];
        *(float4*)&Ws[k * LDS_PITCH + n4 * 4] = v;
    }

    // A-strip staging (512 float4, 2 per thread) into ping-pong buffer.
    auto stage = [&](int t, int bsel) {
        int row0 = (tile0 + t) * 16;
        for (int i = tid; i < 16 * 32; i += 256) {
            int r = i >> 5;
            int c = (i & 31) * 4;
            int g = row0 + r;
            float4 v;
            if (concat_mode) {
                int b = g / NN;
                int n = g - b * NN;
                size_t base = (size_t)(b * NN + n) * 64;
                v = (c < 64) ? *(const float4*)(A0 + base + c)
                             : *(const float4*)(A1 + base + (c - 64));
            } else {
                v = *(const float4*)(A0 + (size_t)g * DD + c);
            }
            *(float4*)&Xs[bsel][r * LDS_PITCH + c] = v;
        }
    };

    stage(0, 0);
    __syncthreads();

    const int wid   = tid >> 5;          // 0..7 -> N tile
    const int lane  = tid & 31;
    const int hl    = lane >> 4;         // lane half
    const int r     = lane & 15;
    const int n     = wid * 16 + r;
    const float bval = (bias != nullptr) ? bias[n] : 0.0f;

    for (int t = 0; t < TPB; ++t) {
        // Prefetch next A strip into the other buffer (overlaps WMMA chain;
        // last read of that buffer was fenced by the previous __syncthreads).
        if (t + 1 < TPB) stage(t + 1, (t + 1) & 1);

        const float* Xb = Xs[t & 1];
        v8f acc = {};
        for (int k0 = 0; k0 < DD; k0 += 4) {
            int ka = k0 + 2 * hl;
            v2f a, b;
            a.x = Xb[r * LDS_PITCH + ka];
            a.y = Xb[r * LDS_PITCH + ka + 1];
            b.x = Ws[ka * LDS_PITCH + n];
            b.y = Ws[(ka + 1) * LDS_PITCH + n];
            acc = __builtin_amdgcn_wmma_f32_16x16x4_f32(
                /*neg_a=*/false, a, /*neg_b=*/false, b,
                /*c_mod=*/(short)0, acc, /*reuse_a=*/false, /*reuse_b=*/false);
        }

        int row0 = (tile0 + t) * 16;
#pragma unroll
        for (int j = 0; j < 8; ++j) {
            int m = j + 8 * hl;
            float v = acc[j] + bval;
            if (act_leaky) v = (v > 0.0f) ? v : LEAKY * v;
            Out[(size_t)(row0 + m) * DD + n] = v;
        }
        if (t + 1 < TPB) __syncthreads();
    }
}

// ---------------------------------------------------------------------------
// K5: conv init with self-loop term: out[g,:] = xw[g,:] * dinv[g]^2
__global__ void k_selfloop(const float* __restrict__ xw,
                           const float* __restrict__ dinv,
                           float* __restrict__ out) {
    long long t = (long long)blockIdx.x * 256 + threadIdx.x;   // BN*32 float4 units
    if (t >= (long long)BN * 32) return;
    size_t g = (size_t)(t >> 5);
    int c = (int)(t & 31) * 4;
    float w = dinv[g] * dinv[g];
    float4 v = *(const float4*)&xw[g * DD + c];
    v.x *= w; v.y *= w; v.z *= w; v.w *= w;
    *(float4*)&out[g * DD + c] = v;
}

// K6: edge gather-scale-scatter. One wave32 per edge, 4 atomicAdds per lane.
__global__ void k_edge_scatter(const long long* __restrict__ ei,
                               const float* __restrict__ xw,
                               const float* __restrict__ dinv,
                               float* __restrict__ out) {
    long long idx = (long long)blockIdx.x * 8 + (threadIdx.x >> 5);
    if (idx >= (long long)BB * EE) return;
    int b = (int)(idx / EE);
    int e = (int)(idx % EE);
    const long long* eb = ei + (size_t)b * 2 * EE;
    int s = (int)eb[e];
    int d = (int)eb[EE + e];
    float w = dinv[b * NN + s] * dinv[b * NN + d];
    int lane = threadIdx.x & 31;
    const float* srow = xw + (size_t)(b * NN + s) * DD;
    float*       drow = out + (size_t)(b * NN + d) * DD;
#pragma unroll
    for (int i = 0; i < 4; ++i) {
        int c = lane + 32 * i;
        atomicAdd(&drow[c], srow[c] * w);
    }
}

// K7: h = relu(conv + bg) + concat(node, act)   (in place on conv)
__global__ void k_residual(float* __restrict__ conv,
                           const float* __restrict__ bg,
                           const float* __restrict__ node,
                           const float* __restrict__ act) {
    long long t = (long long)blockIdx.x * 256 + threadIdx.x;   // BN*32 float4 units
    if (t >= (long long)BN * 32) return;
    size_t g = (size_t)(t >> 5);
    int c = (int)(t & 31) * 4;
    float4 v  = *(float4*)&conv[g * DD + c];
    float4 bb = *(const float4*)&bg[c];
    size_t base = g * 64;
    float4 x = (c < 64) ? *(const float4*)&node[base + c]
                        : *(const float4*)&act[base + (c - 64)];
    v.x = fmaxf(v.x + bb.x, 0.0f) + x.x;
    v.y = fmaxf(v.y + bb.y, 0.0f) + x.y;
    v.z = fmaxf(v.z + bb.z, 0.0f) + x.z;
    v.w = fmaxf(v.w + bb.w, 0.0f) + x.w;
    *(float4*)&conv[g * DD + c] = v;
}

// K10: out[g] = h[g,:] . W3 + b3 ; one wave per row, shuffle reduction.
__global__ void k_final_mv(const float* __restrict__ h,
                           const float* __restrict__ W3,
                           const float* __restrict__ b3,
                           float* __restrict__ out) {
    long long g = (long long)blockIdx.x * 8 + (threadIdx.x >> 5);
    if (g >= BN) return;
    int lane = threadIdx.x & 31;
    float4 hv = ((const float4*)(h + (size_t)g * DD))[lane];
    float4 wv = ((const float4*)W3)[lane];
    float s = hv.x * wv.x + hv.y * wv.y + hv.z * wv.z + hv.w * wv.w;
#pragma unroll
    for (int off = 16; off > 0; off >>= 1) s += __shfl_xor(s, off, 32);
    if (lane == 0) out[g] = s + b3[0];
}

// ---------------------------------------------------------------------------
extern "C" void kernel_launch(void* const* d_in, const int* in_sizes, int n_in,
                              void* d_out, int out_size, void* d_ws, size_t ws_size,
                              hipStream_t stream) {
    const float*     node = (const float*)d_in[0];
    const float*     act  = (const float*)d_in[1];
    const long long* ei   = (const long long*)d_in[2];
    const float*     Wg   = (const float*)d_in[3];
    const float*     bg   = (const float*)d_in[4];
    const float*     W1   = (const float*)d_in[5];
    const float*     b1   = (const float*)d_in[6];
    const float*     W2   = (const float*)d_in[7];
    const float*     b2   = (const float*)d_in[8];
    const float*     W3   = (const float*)d_in[9];
    const float*     b3   = (const float*)d_in[10];
    float* out = (float*)d_out;

    // Workspace layout (all 256B-aligned): dinv | buf0 | buf1 | buf2
    char* ws = (char*)d_ws;
    float* dinv = (float*)ws;                            // BN floats (640000 B)
    float* buf0 = (float*)(ws + (size_t)BN * 4);         // BN*128  (xw, later h2)
    float* buf1 = buf0 + (size_t)BN * DD;                // conv -> h
    float* buf2 = buf1 + (size_t)BN * DD;                // h1

    const dim3 blk(256);
    const int gemm_blocks = BN / (16 * TPB);             // 625

    // 1-3: symmetric normalization coefficients
    k_deg_init <<<(BN + 255) / 256, blk, 0, stream>>>(dinv);
    k_deg_accum<<<((long long)BB * EE + 255) / 256, blk, 0, stream>>>(ei, dinv);
    k_dinv     <<<(BN + 255) / 256, blk, 0, stream>>>(dinv);

    // 4: xw = concat(node,act) @ Wg  -> buf0   (WMMA f32)
    k_gemm128<<<gemm_blocks, blk, 0, stream>>>(node, act, Wg, nullptr, buf0, 1, 0);

    // 5-6: conv = D^-1/2 A D^-1/2 xw  -> buf1
    k_selfloop    <<<(BN * 32) / 256, blk, 0, stream>>>(buf0, dinv, buf1);
    k_edge_scatter<<<((long long)BB * EE + 7) / 8, blk, 0, stream>>>(ei, buf0, dinv, buf1);

    // 7: h = relu(conv + bg) + x   (in place on buf1)
    k_residual<<<(BN * 32) / 256, blk, 0, stream>>>(buf1, bg, node, act);

    // 8: h1 = leaky(h @ W1 + b1)   buf1 -> buf2   (WMMA f32)
    k_gemm128<<<gemm_blocks, blk, 0, stream>>>(buf1, nullptr, W1, b1, buf2, 0, 1);
    // 9: h2 = leaky(h1 @ W2 + b2)  buf2 -> buf0   (WMMA f32)
    k_gemm128<<<gemm_blocks, blk, 0, stream>>>(buf2, nullptr, W2, b2, buf0, 0, 1);

    // 10: out = h2 @ W3 + b3
    k_final_mv<<<(BN + 7) / 8, blk, 0, stream>>>(buf0, W3, b3, out);
}